// Encoder_8254927142919
// MI455X (gfx1250) — compile-verified
//
#include <hip/hip_runtime.h>
#include <hip/hip_bf16.h>

// ---------------------------------------------------------------------------
// GATv2 x2 + PReLU for gfx1250 (MI455X, wave32, WMMA).
//  - Projections: V_WMMA_F32_16X16X4_F32 (exact fp32 == reference numerics);
//    A-tiles staged global->LDS once per block via GLOBAL_LOAD_ASYNC_TO_LDS_B128
//    (ASYNCcnt / s_wait_asynccnt), read back as ds_load_b64 A-fragments.
//  - Edge phase: L2-resident gather/scatter (~80MB working set < 192MB L2).
// ---------------------------------------------------------------------------

typedef float v2f __attribute__((ext_vector_type(2)));
typedef float v8f __attribute__((ext_vector_type(8)));

#define IN_DIM   512
#define HEADS    2
#define CH       64
#define HC       128   // HEADS*CH
#define EDIM     3
#define NEG_SLOPE 0.2f

#define KC       64    // K-chunk staged in LDS per iteration
#define TSTRIDE  68    // padded row stride (272B: 16B-aligned, 4-bank rotate/row)

// Monotone float<->uint encoding so atomicMax(uint) == float max (handles negatives)
__device__ __forceinline__ unsigned f2ord(float f) {
  unsigned u = __float_as_uint(f);
  return (u & 0x80000000u) ? ~u : (u | 0x80000000u);
}
__device__ __forceinline__ float ord2f(unsigned u) {
  return __uint_as_float((u & 0x80000000u) ? (u & 0x7fffffffu) : ~u);
}

// ---------------------------------------------------------------------------
__global__ void k_zero(float* __restrict__ p, size_t n) {
  size_t i = (size_t)blockIdx.x * blockDim.x + threadIdx.x;
  size_t stride = (size_t)gridDim.x * blockDim.x;
  for (; i < n; i += stride) p[i] = 0.0f;
}

// Self-loop "mean" edge attr: segment-sum edge_attr over dst + counts
__global__ void k_sl_accum(const long long* __restrict__ dst,
                           const float* __restrict__ eattr,
                           float* __restrict__ sum, float* __restrict__ cnt,
                           int E) {
  int e = blockIdx.x * blockDim.x + threadIdx.x;
  if (e >= E) return;
  int d = (int)dst[e];
  atomicAdd(&sum[d * 3 + 0], eattr[e * 3 + 0]);
  atomicAdd(&sum[d * 3 + 1], eattr[e * 3 + 1]);
  atomicAdd(&sum[d * 3 + 2], eattr[e * 3 + 2]);
  atomicAdd(&cnt[d], 1.0f);
}

__global__ void k_sl_mean(const float* __restrict__ sum,
                          const float* __restrict__ cnt,
                          float* __restrict__ mean, int N) {
  int n = blockIdx.x * blockDim.x + threadIdx.x;
  if (n >= N) return;
  float c = cnt[n]; c = c < 1.0f ? 1.0f : c;
  float inv = 1.0f / c;
  mean[n * 3 + 0] = sum[n * 3 + 0] * inv;
  mean[n * 3 + 1] = sum[n * 3 + 1] * inv;
  mean[n * 3 + 2] = sum[n * 3 + 2] * inv;
}

// ---------------------------------------------------------------------------
// Fused projection GEMM: Yl = X@Wl + Bl, Yr = X@Wr + Br   (X: [N,K], W: [K,128])
// blockIdx.y selects (l|r). 8 waves/block; wave w owns the 16x16 output tile at
// column tile w. K consumed in LDS-staged chunks of KC, 4 at a time with
// V_WMMA_F32_16X16X4_F32.
// ISA layouts (§7.12.2):
//   A 16x4 f32: lane[0..15] row M=l, K=2*half+{0,1} per VGPR   (half = lane>=16)
//   B 4x16 f32: lane[0..15] col N=l, K rows 2*half+{0,1}
//   D 16x16:    vgpr v, lane l -> row = v + 8*half, col = l%16
__global__ void k_proj_wmma(const float* __restrict__ X,
                            const float* __restrict__ Wl, const float* __restrict__ Bl,
                            const float* __restrict__ Wr, const float* __restrict__ Br,
                            float* __restrict__ Yl, float* __restrict__ Yr,
                            int K) {
  const float* W = (blockIdx.y == 0) ? Wl : Wr;
  const float* B = (blockIdx.y == 0) ? Bl : Br;
  float*       Y = (blockIdx.y == 0) ? Yl : Yr;

  __shared__ float tile[16 * TSTRIDE];

  const int lane = threadIdx.x & 31;
  const int wave = threadIdx.x >> 5;      // 0..7 -> column tile
  const int row0 = blockIdx.x * 16;
  const int col0 = wave * 16;
  const int half = lane >> 4;             // 0 | 1
  const int l16  = lane & 15;

  // staging assignment: thread t copies 16B: row = t/16, float offset = (t%16)*4
  const int srow = threadIdx.x >> 4;
  const int sch  = (threadIdx.x & 15) * 4;
  const float* __restrict__ gsrc = X + (size_t)(row0 + srow) * K + sch;

  // LDS byte offset for the async copy's VDST operand. Deriving it via
  // ptrtoint of &tile[...] (flat LDS pointers carry the wave-relative LDS
  // offset in addr[31:0], ISA §10.2) ESCAPES `tile`, so the asm "memory"
  // clobber legally clobbers it and the ds_load A-fragments below cannot be
  // folded to poison (that was the v[0:1]-overlap miscompile in round 2).
  const unsigned ldsoff =
      (unsigned)(size_t)(&tile[srow * TSTRIDE + sch]);

  const float* __restrict__ wcol = W + col0 + l16;

  v8f acc = {};
  for (int k0 = 0; k0 < K; k0 += KC) {
    // ---- stage A-chunk (16 x KC floats) global -> LDS, one B128 per thread ----
    const float* gp = gsrc + k0;
    __builtin_prefetch(gp + KC, 0, 3);   // pull next chunk toward WGP$
    asm volatile("global_load_async_to_lds_b128 %0, %1, off"
                 :: "v"(ldsoff), "v"(gp) : "memory");
    asm volatile("s_wait_asynccnt 0x0" ::: "memory");
    __syncthreads();

    // ---- consume chunk: KC/4 WMMAs; A from LDS, B (tiny, L2-resident) global --
#pragma unroll 8
    for (int kk = 0; kk < KC; kk += 4) {
      v2f a, b;
      a.x = tile[l16 * TSTRIDE + kk + 2 * half + 0];
      a.y = tile[l16 * TSTRIDE + kk + 2 * half + 1];
      b.x = wcol[(size_t)(k0 + kk + 2 * half + 0) * HC];
      b.y = wcol[(size_t)(k0 + kk + 2 * half + 1) * HC];
      acc = __builtin_amdgcn_wmma_f32_16x16x4_f32(
          /*neg_a=*/false, a, /*neg_b=*/false, b,
          /*c_mod=*/(short)0, acc, /*reuse_a=*/false, /*reuse_b=*/false);
    }
    __syncthreads();   // before next stage overwrites the tile
  }

#pragma unroll
  for (int v = 0; v < 8; ++v) {
    int r = row0 + v + 8 * half;
    int c = col0 + l16;
    Y[(size_t)r * HC + c] = acc[v] + B[c];
  }
}

// ---------------------------------------------------------------------------
// Edge score: one wave32 per edge. lanes 0..15 -> head 0 (c 0..63, 4/lane),
// lanes 16..31 -> head 1. Inline 3x128 edge-attr matvec. Butterfly reduce with
// xor offsets <=8 stays within each 16-lane head group (wave32-aware).
__global__ void k_edge_score(const long long* __restrict__ srcI,
                             const long long* __restrict__ dstI,
                             const float* __restrict__ eattr,
                             const float* __restrict__ mean_attr,
                             const float* __restrict__ xl,
                             const float* __restrict__ xr,
                             const float* __restrict__ we,   // [3,128]
                             const float* __restrict__ att,  // [H*C]=128
                             float* __restrict__ score,      // [E',2]
                             unsigned* __restrict__ smax,    // [N,2] ordered-uint
                             int EP, int E) {
  int e = blockIdx.x * (blockDim.x >> 5) + (threadIdx.x >> 5);
  if (e >= EP) return;
  const int lane = threadIdx.x & 31;

  int s, d; float a0, a1, a2;
  if (e < E) {
    s = (int)srcI[e]; d = (int)dstI[e];
    a0 = eattr[e * 3 + 0]; a1 = eattr[e * 3 + 1]; a2 = eattr[e * 3 + 2];
  } else {
    s = d = e - E;  // self-loop, mean attr
    a0 = mean_attr[s * 3 + 0]; a1 = mean_attr[s * 3 + 1]; a2 = mean_attr[s * 3 + 2];
  }

  const int i0 = lane * 4;
  const float* __restrict__ xls = xl + (size_t)s * HC + i0;
  const float* __restrict__ xrd = xr + (size_t)d * HC + i0;

  float partial = 0.0f;
#pragma unroll
  for (int j = 0; j < 4; ++j) {
    int i = i0 + j;
    float ee = a0 * we[i] + a1 * we[HC + i] + a2 * we[2 * HC + i];
    float m  = xls[j] + xrd[j] + ee;
    float lr = (m >= 0.0f) ? m : NEG_SLOPE * m;
    partial += lr * att[i];
  }
#pragma unroll
  for (int off = 8; off > 0; off >>= 1)
    partial += __shfl_xor(partial, off, 32);

  if ((lane & 15) == 0) {
    int h = lane >> 4;
    score[(size_t)e * 2 + h] = partial;
    atomicMax(&smax[(size_t)d * 2 + h], f2ord(partial));
  }
}

// alpha = exp(score - smax[dst]); denom[dst] += alpha (alpha overwrites score)
__global__ void k_edge_alpha(const long long* __restrict__ dstI,
                             float* __restrict__ score,
                             const unsigned* __restrict__ smax,
                             float* __restrict__ denom,
                             int EP, int E) {
  int t = blockIdx.x * blockDim.x + threadIdx.x;
  if (t >= EP * 2) return;
  int e = t >> 1, h = t & 1;
  int d = (e < E) ? (int)dstI[e] : (e - E);
  float m = ord2f(smax[(size_t)d * 2 + h]);
  float a = __expf(score[t] - m);
  score[t] = a;
  atomicAdd(&denom[(size_t)d * 2 + h], a);
}

// accum[dst] += (alpha/denom[dst]) * xl[src]  — one wave per edge, 4 ch/lane
__global__ void k_edge_accum(const long long* __restrict__ srcI,
                             const long long* __restrict__ dstI,
                             const float* __restrict__ xl,
                             const float* __restrict__ alpha,
                             const float* __restrict__ denom,
                             float* __restrict__ accum,
                             int EP, int E) {
  int e = blockIdx.x * (blockDim.x >> 5) + (threadIdx.x >> 5);
  if (e >= EP) return;
  const int lane = threadIdx.x & 31;

  int s, d;
  if (e < E) { s = (int)srcI[e]; d = (int)dstI[e]; }
  else       { s = d = e - E; }

  int h = lane >> 4;
  float coef = alpha[(size_t)e * 2 + h] / denom[(size_t)d * 2 + h];
  int i0 = lane * 4;
  const float* __restrict__ xs = xl + (size_t)s * HC + i0;
  float* __restrict__ ad = accum + (size_t)d * HC + i0;
#pragma unroll
  for (int j = 0; j < 4; ++j)
    atomicAdd(&ad[j], coef * xs[j]);
}

// out[n,c] = mean over heads + bias  (optional PReLU for the final layer)
__global__ void k_finalize(const float* __restrict__ accum,
                           const float* __restrict__ bo,
                           const float* __restrict__ prelu,  // nullptr -> skip
                           float* __restrict__ out, int N) {
  int t = blockIdx.x * blockDim.x + threadIdx.x;
  if (t >= N * CH) return;
  int n = t / CH, c = t - n * CH;
  float v = 0.5f * (accum[(size_t)n * HC + c] + accum[(size_t)n * HC + CH + c]) + bo[c];
  if (prelu) v = (v >= 0.0f) ? v : prelu[c] * v;
  out[t] = v;
}

// ---------------------------------------------------------------------------
extern "C" void kernel_launch(void* const* d_in, const int* in_sizes, int n_in,
                              void* d_out, int out_size, void* d_ws, size_t ws_size,
                              hipStream_t stream) {
  const float*     x     = (const float*)d_in[0];
  const long long* ei    = (const long long*)d_in[1];   // int64 [2,E]
  const float*     eattr = (const float*)d_in[2];
  const float *wl1 = (const float*)d_in[3],  *bl1 = (const float*)d_in[4];
  const float *wr1 = (const float*)d_in[5],  *br1 = (const float*)d_in[6];
  const float *we1 = (const float*)d_in[7],  *att1= (const float*)d_in[8];
  const float *bo1 = (const float*)d_in[9];
  const float *wl2 = (const float*)d_in[10], *bl2 = (const float*)d_in[11];
  const float *wr2 = (const float*)d_in[12], *br2 = (const float*)d_in[13];
  const float *we2 = (const float*)d_in[14], *att2= (const float*)d_in[15];
  const float *bo2 = (const float*)d_in[16];
  const float *prw = (const float*)d_in[17];

  const int N  = in_sizes[0] / IN_DIM;
  const int E  = in_sizes[1] / 2;
  const int EP = E + N;
  const long long* srcI = ei;
  const long long* dstI = ei + E;

  // -------- workspace layout (floats) --------
  float* ws = (float*)d_ws;
  size_t off = 0;
  float* sl_sum = ws + off; off += (size_t)N * 3;
  float* sl_cnt = ws + off; off += (size_t)N;        // contiguous with sl_sum
  float* meanA  = ws + off; off += (size_t)N * 3;
  float* xl     = ws + off; off += (size_t)N * HC;
  float* xr     = ws + off; off += (size_t)N * HC;
  float* score  = ws + off; off += (size_t)EP * 2;   // alpha in-place
  float* smaxF  = ws + off; off += (size_t)N * 2;    // ordered-uint, zero-init OK
  float* denom  = ws + off; off += (size_t)N * 2;    // contiguous with smax, accum
  float* accum  = ws + off; off += (size_t)N * HC;
  float* h1     = ws + off; off += (size_t)N * CH;
  if (ws_size < off * sizeof(float)) return;         // insufficient scratch

  unsigned* smax = (unsigned*)smaxF;
  float* outF = (float*)d_out;

  const int ZB = 2048, TB = 256;
  // -------- self-loop mean edge attr --------
  k_zero<<<ZB, TB, 0, stream>>>(sl_sum, (size_t)N * 4);          // sums + counts
  k_sl_accum<<<(E + TB - 1) / TB, TB, 0, stream>>>(dstI, eattr, sl_sum, sl_cnt, E);
  k_sl_mean<<<(N + TB - 1) / TB, TB, 0, stream>>>(sl_sum, sl_cnt, meanA, N);

  for (int layer = 0; layer < 2; ++layer) {
    const float* X   = (layer == 0) ? x   : h1;
    const int    K   = (layer == 0) ? IN_DIM : CH;
    const float* wl  = (layer == 0) ? wl1 : wl2;
    const float* bl  = (layer == 0) ? bl1 : bl2;
    const float* wr  = (layer == 0) ? wr1 : wr2;
    const float* br  = (layer == 0) ? br1 : br2;
    const float* we  = (layer == 0) ? we1 : we2;
    const float* att = (layer == 0) ? att1 : att2;
    const float* bo  = (layer == 0) ? bo1 : bo2;
    float* dst_out   = (layer == 0) ? h1 : outF;
    const float* pre = (layer == 0) ? nullptr : prw;

    // reset smax | denom | accum (contiguous -> single zero pass)
    k_zero<<<ZB, TB, 0, stream>>>(smaxF, (size_t)N * (2 + 2 + HC));

    // fp32 WMMA projections (LDS-staged A): grid.y = {xl, xr}
    dim3 pg(N / 16, 2);
    k_proj_wmma<<<pg, 256, 0, stream>>>(X, wl, bl, wr, br, xl, xr, K);

    const int EW = (EP + 7) / 8;   // 8 waves (edges) per 256-thread block
    k_edge_score<<<EW, 256, 0, stream>>>(srcI, dstI, eattr, meanA, xl, xr,
                                         we, att, score, smax, EP, E);
    k_edge_alpha<<<(EP * 2 + TB - 1) / TB, TB, 0, stream>>>(dstI, score, smax,
                                                            denom, EP, E);
    k_edge_accum<<<EW, 256, 0, stream>>>(srcI, dstI, xl, score, denom, accum,
                                         EP, E);
    k_finalize<<<(N * CH + TB - 1) / TB, TB, 0, stream>>>(accum, bo, pre,
                                                          dst_out, N);
  }
}